// CRFLayer_24129126269516
// MI455X (gfx1250) — compile-verified
//
#include <hip/hip_runtime.h>

#define CRF_B 256
#define CRF_T 2048
#define CRF_N 64
#define CRF_S 128                      // edges per backward segment (power of 2)
#define CRF_NSEG ((CRF_T - 1 + CRF_S - 1) / CRF_S)   // 16

#if defined(__has_builtin)
#if __has_builtin(__builtin_amdgcn_tensor_load_to_lds) && __has_builtin(__builtin_amdgcn_s_wait_tensorcnt)
#define CRF_HAVE_TDM 1
#endif
#endif

typedef unsigned int u32x4 __attribute__((ext_vector_type(4)));
typedef int i32x4 __attribute__((ext_vector_type(4)));
typedef int i32x8 __attribute__((ext_vector_type(8)));

// One workgroup per batch element. 256 threads: thread tid handles column
// j = tid>>2, i-chunk h = tid&3 (16 entries). trans sub-column lives in 16
// VGPRs; alpha double-buffered in LDS; backpointers (2047x64 u8 = 131KB) in
// LDS so the backward chase never touches global memory. A path-compression
// table (reach/ckpt) is maintained in the forward pass so the backward chase
// is 16 parallel 128-hop walks instead of one serial 2047-hop walk.
__global__ __launch_bounds__(256) void crf_viterbi_fused(
    const float* __restrict__ x,
    const int* __restrict__ seq_len,
    const float* __restrict__ trans,
    float* __restrict__ out)
{
  __shared__ float s_alpha[2][CRF_N];
  __shared__ unsigned char s_bp[(CRF_T - 1) * CRF_N];
  __shared__ unsigned char s_reach[2][CRF_N];
  __shared__ unsigned char s_ckpt[CRF_NSEG][CRF_N];
  __shared__ int s_bstate[CRF_NSEG + 1];
#if CRF_HAVE_TDM
  __shared__ float s_trans[CRF_N * CRF_N];
#endif
  __shared__ int s_last;

  const int b   = blockIdx.x;
  const int tid = threadIdx.x;
  const int j   = tid >> 2;
  const int h   = tid & 3;
  const int sl  = seq_len[b];

  const float* xb   = x + (size_t)b * CRF_T * CRF_N;
  float* tags_out   = out + (size_t)b * CRF_T;
  float* scores_out = out + (size_t)CRF_B * CRF_T;

#if CRF_HAVE_TDM
  // Stage trans (64x64 f32, 16KB) into LDS with one Tensor-DMA op from wave 0.
  if (tid < 32) {
    unsigned lds_off = (unsigned)(uintptr_t)(void*)s_trans;   // generic low 32 bits == LDS offset
    unsigned long long ga = (unsigned long long)(uintptr_t)trans;
    u32x4 g0;
    g0.x = 1u;                                         // count=1 valid descriptor
    g0.y = lds_off;                                    // lds_addr [63:32]
    g0.z = (unsigned)(ga & 0xFFFFFFFFull);             // global_addr [95:64]
    g0.w = (unsigned)((ga >> 32) & 0x1FFFFFFull)       // global_addr [120:96]
         | (2u << 30);                                 // type=2 ("image")
    i32x8 g1;
    g1.s0 = (2 << 16);          // data_size = 4 bytes, wg_mask = 0
    g1.s1 = (CRF_N << 16);      // tensor_dim0 = 64 (bits 79:48, low half)
    g1.s2 = (CRF_N << 16);      // tensor_dim1 = 64 (bits 111:80, low half)
    g1.s3 = (CRF_N << 16);      // tile_dim0   = 64 (bits 127:112)
    g1.s4 = CRF_N;              // tile_dim1   = 64, tile_dim2 = 0
    g1.s5 = CRF_N;              // tensor_dim0_stride = 64 (low 32)
    g1.s6 = 0;
    g1.s7 = 0;
    i32x4 gz4 = {0, 0, 0, 0};       // 2D tensor: groups 2/3 unused
    i32x8 gz8 = {0, 0, 0, 0, 0, 0, 0, 0};
    __builtin_amdgcn_tensor_load_to_lds(g0, g1, gz4, gz4, gz8, 0);
  }
  __builtin_amdgcn_s_wait_tensorcnt(0);
#endif

  if (tid < CRF_N) s_alpha[0][tid] = __builtin_nontemporal_load(xb + tid); // alpha0 = x[b,0,:]
  __syncthreads();

  // trans sub-column in registers: tc[k] = trans[16h+k][j]
  float tc[16];
#pragma unroll
  for (int k = 0; k < 16; ++k) {
#if CRF_HAVE_TDM
    tc[k] = s_trans[(h * 16 + k) * CRF_N + j];
#else
    tc[k] = trans[(h * 16 + k) * CRF_N + j];
#endif
  }

  int cur = 0;
  float xv = __builtin_nontemporal_load(xb + CRF_N + j);    // prefetch x[b,1,j]
  for (int t = 1; t < sl; ++t) {
    const float xcur = xv;
    const int tn = (t + 1 < sl) ? (t + 1) : t;              // clamped prefetch index
    xv = __builtin_nontemporal_load(xb + (size_t)tn * CRF_N + j);

    float av[16];
    const float4* ap = (const float4*)(&s_alpha[cur][h * 16]);
    *(float4*)(&av[0])  = ap[0];
    *(float4*)(&av[4])  = ap[1];
    *(float4*)(&av[8])  = ap[2];
    *(float4*)(&av[12]) = ap[3];

    float best = av[0] + tc[0];
    int argi = h * 16;
#pragma unroll
    for (int k = 1; k < 16; ++k) {
      float v = av[k] + tc[k];
      bool gt = v > best;                       // strict > keeps first index on ties
      best = gt ? v : best;
      argi = gt ? (h * 16 + k) : argi;
    }
    // combine the 4 partials of each quad inside the wave (lanes xor 1, xor 2)
#pragma unroll
    for (int m = 1; m <= 2; m <<= 1) {
      float ov = __shfl_xor(best, m, 32);
      int   oi = __shfl_xor(argi, m, 32);
      bool take = (ov > best) || ((ov == best) && (oi < argi));
      best = take ? ov : best;
      argi = take ? oi : argi;
    }
    if (h == 0) {
      const int e = t - 1;                      // edge index
      s_alpha[cur ^ 1][j] = best + xcur;
      s_bp[e * CRF_N + j] = (unsigned char)argi;
      // path compression: state at current segment start for a path at j now
      int rnew = ((e & (CRF_S - 1)) == 0) ? argi : (int)s_reach[cur][argi];
      s_reach[cur ^ 1][j] = (unsigned char)rnew;
      if (((e & (CRF_S - 1)) == (CRF_S - 1)) || (t == sl - 1))
        s_ckpt[e / CRF_S][j] = (unsigned char)rnew;   // checkpoint at segment end
    }
    cur ^= 1;
    __syncthreads();
  }

  // scores / last = max/argmax over final alpha (wave 0 only)
  if (tid < 32) {
    float v0 = s_alpha[cur][tid];
    float v1 = s_alpha[cur][tid + 32];
    float mv = v0; int mi = tid;
    { bool gt = v1 > mv; mv = gt ? v1 : mv; mi = gt ? tid + 32 : mi; }
#pragma unroll
    for (int m = 1; m <= 16; m <<= 1) {
      float ov = __shfl_xor(mv, m, 32);
      int   oi = __shfl_xor(mi, m, 32);
      bool take = (ov > mv) || ((ov == mv) && (oi < mi));
      mv = take ? ov : mv;
      mi = take ? oi : mi;
    }
    if (tid == 0) { s_last = mi; scores_out[b] = mv; }
  }
  __syncthreads();
  const int last = s_last;

  // t >= sl: identity backpointers => tag stays `last`. Coalesced parallel fill.
  for (int t = sl + tid; t < CRF_T; t += 256) tags_out[t] = (float)last;

  const int ne = sl - 1;                       // number of valid edges
  if (ne > 0) {
    const int nseg = (ne + CRF_S - 1) / CRF_S;
    // Boundary states via checkpoint chain (<=16 dependent hops).
    if (tid == 0) {
      int st = last;                           // state at time ne
      for (int c = nseg - 1; c >= 0; --c) {
        st = s_ckpt[c][st];                    // state at time c*CRF_S
        s_bstate[c] = st;
      }
    }
    __syncthreads();
    // Parallel interior walks: thread c walks segment c (<=128 dependent hops).
    if (tid < nseg) {
      const int c = tid;
      const int eEnd = ((c + 1) * CRF_S < ne) ? (c + 1) * CRF_S : ne;  // exclusive
      int st = (c == nseg - 1) ? last : s_bstate[c + 1];               // state at time eEnd
      for (int e = eEnd - 1; e >= c * CRF_S; --e) {
        tags_out[e + 1] = (float)st;
        st = s_bp[e * CRF_N + st];
      }
      if (c == 0) tags_out[0] = (float)st;
    }
  } else {
    if (tid == 0) tags_out[0] = (float)last;
  }
}

extern "C" void kernel_launch(void* const* d_in, const int* in_sizes, int n_in,
                              void* d_out, int out_size, void* d_ws, size_t ws_size,
                              hipStream_t stream) {
  const float* x       = (const float*)d_in[0];
  const int*   seq_len = (const int*)d_in[1];
  const float* trans   = (const float*)d_in[2];
  float* out = (float*)d_out;
  (void)in_sizes; (void)n_in; (void)out_size; (void)d_ws; (void)ws_size;
  crf_viterbi_fused<<<CRF_B, 256, 0, stream>>>(x, seq_len, trans, out);
}